// SegmentMambaEmbed_76020921140205
// MI455X (gfx1250) — compile-verified
//
#include <hip/hip_runtime.h>
#include <hip/hip_bf16.h>
#include <math.h>

// Problem constants (from reference)
#define B_      8
#define NSEG_   16
#define LSEQ    128
#define DM_     128
#define DI_     256
#define NST     16
#define DCONV_  4
#define DTR_    8
#define NL_     3
#define DOUT_   64
#define BEFF    (B_ * NSEG_)          // 128 sequences
#define MROWS   (BEFF * LSEQ)         // 16384 rows for all GEMMs
#define NXP_PAD 48                    // 40 (DTR+2N) padded to 3 WMMA tiles

typedef float v2f __attribute__((ext_vector_type(2)));
typedef float v8f __attribute__((ext_vector_type(8)));

// ---------------------------------------------------------------------------
// WMMA fp32 GEMM:  C[M, Npad] = A[M, lda(rows m,k)] * W[Npad, ldw(rows n,k)]^T
// One wave computes one 16x16 tile via V_WMMA_F32_16X16X4_F32, K in steps of 4.
// A fragment (16x4): lane l<16 -> (M=l, K=k0+{0,1}); lane>=16 -> K=k0+{2,3}.
// B fragment (4x16): lane l<16 -> (N=l, K=k0+{0,1}); lane>=16 -> K=k0+{2,3}.
// Both are contiguous 8B loads since k is the fastest axis (row-major, NT).
// EPI==1: fused bias + softplus (for the dt projection).
// ---------------------------------------------------------------------------
template <int EPI>
__global__ __launch_bounds__(256) void wmma_gemm_nt(
    const float* __restrict__ A, int lda,
    const float* __restrict__ W, int ldw,
    float* __restrict__ C, int ldc,
    const float* __restrict__ bias,
    int Npad, int K)
{
    const int wave = threadIdx.x >> 5;      // 8 waves per block (wave32)
    const int lane = threadIdx.x & 31;
    const int l16  = lane & 15;
    const int hi   = lane >> 4;             // 0 or 1 (selects K pair)
    const int tile_m = blockIdx.x * 16;
    const int tile_n = (blockIdx.y * 8 + wave) * 16;
    if (tile_n >= Npad) return;             // wave-uniform: EXEC stays all-1s

    const float* __restrict__ Arow = A + (size_t)(tile_m + l16) * lda + 2 * hi;
    const float* __restrict__ Wrow = W + (size_t)(tile_n + l16) * ldw + 2 * hi;

    v8f acc = {};
#pragma unroll 4
    for (int k0 = 0; k0 < K; k0 += 4) {
        v2f a = *(const v2f*)(Arow + k0);
        v2f b = *(const v2f*)(Wrow + k0);
        // (neg_a, A, neg_b, B, c_mod, C, reuse_a, reuse_b)
        acc = __builtin_amdgcn_wmma_f32_16x16x4_f32(
            false, a, false, b, (short)0, acc, false, false);
    }

    float bval = 0.0f;
    if (EPI == 1) bval = bias[tile_n + l16];

    // C/D layout: VGPR g -> row (g + 8*hi), col l16
    float* __restrict__ Crow = C + (size_t)tile_m * ldc + tile_n + l16;
#pragma unroll
    for (int g = 0; g < 8; ++g) {
        float v = acc[g];
        if (EPI == 1) {
            v += bval;
            v = (v > 20.0f) ? v : log1pf(__expf(v));   // softplus
        }
        Crow[(size_t)(g + 8 * hi) * ldc] = v;
    }
}

// ---------------------------------------------------------------------------
// Depthwise causal conv (DCONV=4) + bias + SiLU over the first DI channels of
// xz; one block per row (b,l), one thread per channel.
// ---------------------------------------------------------------------------
__global__ __launch_bounds__(256) void conv_silu_kernel(
    const float* __restrict__ xz,      // [MROWS, 2*DI]
    const float* __restrict__ cw,      // [DI, DCONV]
    const float* __restrict__ cb,      // [DI]
    float* __restrict__ xc)            // [MROWS, DI]
{
    const int r = blockIdx.x;          // b*LSEQ + l
    const int c = threadIdx.x;         // channel
    const int l = r & (LSEQ - 1);

    float acc = cb[c];
#pragma unroll
    for (int k = 0; k < DCONV_; ++k) {
        const int ls = l - (DCONV_ - 1) + k;
        if (ls >= 0)
            acc += xz[(size_t)(r - (DCONV_ - 1) + k) * (2 * DI_) + c] *
                   cw[c * DCONV_ + k];
    }
    xc[(size_t)r * DI_ + c] = acc / (1.0f + __expf(-acc));   // silu
}

// Pad xproj weights (40,256) -> (48,256) with zero rows so WMMA tiles are full.
__global__ __launch_bounds__(256) void pad_xproj_kernel(
    const float* __restrict__ w, float* __restrict__ wp)
{
    const int n = blockIdx.x;          // 0..47
    const int k = threadIdx.x;         // 0..255
    wp[(size_t)n * DI_ + k] = (n < (DTR_ + 2 * NST)) ? w[(size_t)n * DI_ + k] : 0.0f;
}

// ---------------------------------------------------------------------------
// Selective scan: one block per sequence, one thread per channel d.
// h[NST] lives in registers; per-step B/C (shared by all d) broadcast via LDS.
// Fuses the D*x skip and the silu(z) gate, writing y for the out_proj GEMM.
// ---------------------------------------------------------------------------
__global__ __launch_bounds__(256) void scan_kernel(
    const float* __restrict__ dbl,     // [MROWS, NXP_PAD]: [0:8)=dt_raw, [8:24)=B, [24:40)=C
    const float* __restrict__ dtb,     // [MROWS, DI] softplus(dt)
    const float* __restrict__ xc,      // [MROWS, DI]
    const float* __restrict__ xz,      // [MROWS, 2*DI] (z = cols [DI, 2*DI))
    const float* __restrict__ A_log,   // [DI, NST]
    const float* __restrict__ Dp,      // [DI]
    float* __restrict__ yb)            // [MROWS, DI]
{
    const int b = blockIdx.x;          // sequence
    const int d = threadIdx.x;         // channel

    float Arow[NST];
#pragma unroll
    for (int n = 0; n < NST; ++n) Arow[n] = -__expf(A_log[d * NST + n]);
    const float Dv = Dp[d];

    float h[NST];
#pragma unroll
    for (int n = 0; n < NST; ++n) h[n] = 0.0f;

    __shared__ float sB[NST];
    __shared__ float sC[NST];

    for (int l = 0; l < LSEQ; ++l) {
        const int r = b * LSEQ + l;
        __syncthreads();
        if (d < NST)            sB[d]        = dbl[(size_t)r * NXP_PAD + DTR_ + d];
        else if (d < 2 * NST)   sC[d - NST]  = dbl[(size_t)r * NXP_PAD + DTR_ + NST + (d - NST)];
        __syncthreads();

        const float dt  = dtb[(size_t)r * DI_ + d];
        const float x   = xc [(size_t)r * DI_ + d];
        const float dtx = dt * x;
        float y = 0.0f;
#pragma unroll
        for (int n = 0; n < NST; ++n) {
            h[n] = __expf(dt * Arow[n]) * h[n] + dtx * sB[n];
            y   += h[n] * sC[n];
        }
        const float z = xz[(size_t)r * (2 * DI_) + DI_ + d];
        const float g = z / (1.0f + __expf(-z));           // silu(z)
        yb[(size_t)r * DI_ + d] = (y + Dv * x) * g;
    }
}

// Mean-pool over L, final 128->64 projection, tanh. One block per sequence.
__global__ __launch_bounds__(128) void pool_proj_tanh_kernel(
    const float* __restrict__ u,       // [MROWS, DM]
    const float* __restrict__ pw,      // [DOUT, DM]
    const float* __restrict__ pb,      // [DOUT]
    float* __restrict__ out)           // [BEFF, DOUT]
{
    __shared__ float pooled[DM_];
    const int b = blockIdx.x;
    const int t = threadIdx.x;         // 128 threads

    float s = 0.0f;
    for (int l = 0; l < LSEQ; ++l) s += u[(size_t)(b * LSEQ + l) * DM_ + t];
    pooled[t] = s * (1.0f / LSEQ);
    __syncthreads();

    if (t < DOUT_) {
        float acc = pb[t];
        for (int d = 0; d < DM_; ++d) acc += pooled[d] * pw[t * DM_ + d];
        out[b * DOUT_ + t] = tanhf(acc);
    }
}

// ---------------------------------------------------------------------------
static inline void launch_gemm(const float* A, int lda, const float* W, int ldw,
                               float* C, int ldc, const float* bias,
                               int Npad, int K, int epi, hipStream_t stream)
{
    dim3 grid(MROWS / 16, (Npad / 16 + 7) / 8);
    dim3 block(256);
    if (epi == 0)
        wmma_gemm_nt<0><<<grid, block, 0, stream>>>(A, lda, W, ldw, C, ldc, bias, Npad, K);
    else
        wmma_gemm_nt<1><<<grid, block, 0, stream>>>(A, lda, W, ldw, C, ldc, bias, Npad, K);
}

extern "C" void kernel_launch(void* const* d_in, const int* in_sizes, int n_in,
                              void* d_out, int out_size, void* d_ws, size_t ws_size,
                              hipStream_t stream)
{
    (void)in_sizes; (void)n_in; (void)out_size; (void)ws_size;

    const float* x        = (const float*)d_in[0];   // (B,NSEG,L,DM) == (MROWS, DM)
    const float* in_w     = (const float*)d_in[1];   // (NL, 2*DI, DM)
    const float* conv_w   = (const float*)d_in[2];   // (NL, DI, DCONV)
    const float* conv_b   = (const float*)d_in[3];   // (NL, DI)
    const float* xproj_w  = (const float*)d_in[4];   // (NL, 40, DI)
    const float* dt_w     = (const float*)d_in[5];   // (NL, DI, DTR)
    const float* dt_b     = (const float*)d_in[6];   // (NL, DI)
    const float* A_log    = (const float*)d_in[7];   // (NL, DI, N)
    const float* Dp       = (const float*)d_in[8];   // (NL, DI)
    const float* out_w    = (const float*)d_in[9];   // (NL, DM, DI)
    const float* proj_w   = (const float*)d_in[10];  // (DOUT, DM)
    const float* proj_b   = (const float*)d_in[11];  // (DOUT,)
    float* out = (float*)d_out;

    // Workspace layout (floats), ~104 MB total
    float* ws  = (float*)d_ws;
    float* u1  = ws;                                  // MROWS*DM
    float* u2  = u1  + (size_t)MROWS * DM_;           // MROWS*DM
    float* xz  = u2  + (size_t)MROWS * DM_;           // MROWS*2*DI
    float* xc  = xz  + (size_t)MROWS * 2 * DI_;       // MROWS*DI
    float* dbl = xc  + (size_t)MROWS * DI_;           // MROWS*NXP_PAD
    float* dtb = dbl + (size_t)MROWS * NXP_PAD;       // MROWS*DI
    float* yb  = dtb + (size_t)MROWS * DI_;           // MROWS*DI
    float* wxp = yb  + (size_t)MROWS * DI_;           // NXP_PAD*DI

    const float* uin = x;
    float* uout = u1;

    for (int i = 0; i < NL_; ++i) {
        const float* in_w_i   = in_w    + (size_t)i * 2 * DI_ * DM_;
        const float* conv_w_i = conv_w  + (size_t)i * DI_ * DCONV_;
        const float* conv_b_i = conv_b  + (size_t)i * DI_;
        const float* xproj_i  = xproj_w + (size_t)i * (DTR_ + 2 * NST) * DI_;
        const float* dt_w_i   = dt_w    + (size_t)i * DI_ * DTR_;
        const float* dt_b_i   = dt_b    + (size_t)i * DI_;
        const float* A_log_i  = A_log   + (size_t)i * DI_ * NST;
        const float* Dp_i     = Dp      + (size_t)i * DI_;
        const float* out_w_i  = out_w   + (size_t)i * DM_ * DI_;

        // 1) in_proj: xz = u @ in_w^T   (M=16384, K=128, N=512)
        launch_gemm(uin, DM_, in_w_i, DM_, xz, 2 * DI_, nullptr, 2 * DI_, DM_, 0, stream);

        // 2) depthwise causal conv + SiLU
        conv_silu_kernel<<<MROWS, DI_, 0, stream>>>(xz, conv_w_i, conv_b_i, xc);

        // 3) xproj (N padded 40->48): dbl = xc @ xproj^T  (K=256)
        pad_xproj_kernel<<<NXP_PAD, DI_, 0, stream>>>(xproj_i, wxp);
        launch_gemm(xc, DI_, wxp, DI_, dbl, NXP_PAD, nullptr, NXP_PAD, DI_, 0, stream);

        // 4) dt = softplus(dt_raw @ dt_w^T + dt_b)  (K=8, N=256, fused epilogue)
        launch_gemm(dbl, NXP_PAD, dt_w_i, DTR_, dtb, DI_, dt_b_i, DI_, DTR_, 1, stream);

        // 5) selective scan + skip + gate
        scan_kernel<<<BEFF, DI_, 0, stream>>>(dbl, dtb, xc, xz, A_log_i, Dp_i, yb);

        // 6) out_proj: u_next = y @ out_w^T  (M=16384, K=256, N=128)
        launch_gemm(yb, DI_, out_w_i, DI_, uout, DM_, nullptr, DM_, DI_, 0, stream);

        uin  = uout;
        uout = (uout == u1) ? u2 : u1;
    }

    // 7) mean-pool + final projection + tanh
    pool_proj_tanh_kernel<<<BEFF, DM_, 0, stream>>>(uin, proj_w, proj_b, out);
}